// GCN_2748779070162
// MI455X (gfx1250) — compile-verified
//
#include <hip/hip_runtime.h>
#include <hip/hip_bf16.h>

typedef __attribute__((ext_vector_type(2))) float v2f;
typedef __attribute__((ext_vector_type(8))) float v8f;

#define TPB 256

// ---------------- degree / normalization ----------------

__global__ void deg_init_kernel(float* __restrict__ deg, int n) {
    int i = blockIdx.x * blockDim.x + threadIdx.x;
    if (i < n) deg[i] = 1.0f;  // self-loop contributes 1 to every node's degree
}

__global__ void deg_count_kernel(const int* __restrict__ dst, float* __restrict__ deg, int e) {
    int i = blockIdx.x * blockDim.x + threadIdx.x;
    if (i < e) atomicAdd(&deg[dst[i]], 1.0f);
}

__global__ void deg_rsqrt_kernel(float* __restrict__ deg, int n) {
    int i = blockIdx.x * blockDim.x + threadIdx.x;
    if (i < n) deg[i] = rsqrtf(deg[i]);  // deg >= 1 always (self-loops)
}

// per-edge norm = dinv[src] * dinv[dst], reused by all 4 layers
__global__ void edge_norm_kernel(const int* __restrict__ src, const int* __restrict__ dst,
                                 const float* __restrict__ dinv, float* __restrict__ norm, int e) {
    int i = blockIdx.x * blockDim.x + threadIdx.x;
    if (i < e) norm[i] = dinv[src[i]] * dinv[dst[i]];
}

// ---------------- dense transform: Out[N x F] = A[N x K] @ W[K x F] ----------------
// One wave32 per 16x16 output tile, V_WMMA_F32_16X16X4_F32, K fully unrolled.
// Requires F % 16 == 0; handles only full 16-row tiles (tail kernel covers the rest).

template <int K, int F>
__global__ void gemm_wmma_f32_kernel(const float* __restrict__ A,
                                     const float* __restrict__ W,
                                     float* __restrict__ Out,
                                     int rowTiles) {
    const int wave = (blockIdx.x * blockDim.x + threadIdx.x) >> 5;
    const int lane = threadIdx.x & 31;
    constexpr int nTilesF = F / 16;
    const int rowTile = wave / nTilesF;
    const int colTile = wave - rowTile * nTilesF;
    if (rowTile >= rowTiles) return;  // wave-uniform exit

    const int halfSel = lane >> 4;     // 0: lanes 0-15 (K=k0,k0+1), 1: lanes 16-31 (K=k0+2,k0+3)
    const int laneLo  = lane & 15;
    const long long r = (long long)rowTile * 16 + laneLo;  // A row (M) for this lane
    const int c       = colTile * 16 + laneLo;             // B/D column (N) for this lane
    const float* arow = A + r * K + (halfSel << 1);

    v8f acc = {};
    constexpr int KF = K & ~3;  // full K-steps
#pragma unroll
    for (int k0 = 0; k0 < KF; k0 += 4) {
        v2f a = *(const v2f*)(arow + k0);               // one global_load_b64
        const int ka = k0 + (halfSel << 1);
        v2f b;
        b.x = W[ka * F + c];
        b.y = W[(ka + 1) * F + c];
        acc = __builtin_amdgcn_wmma_f32_16x16x4_f32(false, a, false, b,
                                                    (short)0, acc, false, false);
    }
    if constexpr ((K & 3) != 0) {  // single compile-time remainder step (K=54 -> lanes 16-31 padded)
        const int ka = KF + (halfSel << 1);
        v2f a, b;
        a.x = (ka     < K) ? arow[KF]     : 0.0f;
        a.y = (ka + 1 < K) ? arow[KF + 1] : 0.0f;
        b.x = (ka     < K) ? W[ka * F + c]       : 0.0f;
        b.y = (ka + 1 < K) ? W[(ka + 1) * F + c] : 0.0f;
        acc = __builtin_amdgcn_wmma_f32_16x16x4_f32(false, a, false, b,
                                                    (short)0, acc, false, false);
    }

    // D: VGPR j holds row j (lanes 0-15) / row j+8 (lanes 16-31)
    float* orow = Out + ((long long)rowTile * 16 + (halfSel << 3)) * F + c;
#pragma unroll
    for (int j = 0; j < 8; ++j) orow[(long long)j * F] = acc[j];
}

// scalar fallback for trailing rows (nrows % 16) — empty at N=100000
__global__ void gemm_tail_kernel(const float* __restrict__ A, const float* __restrict__ W,
                                 float* __restrict__ Out, int rowStart, int nrows, int K, int F) {
    int t = blockIdx.x * blockDim.x + threadIdx.x;
    int total = (nrows - rowStart) * F;
    if (t >= total) return;
    int r = rowStart + t / F;
    int c = t % F;
    float s = 0.0f;
    for (int k = 0; k < K; ++k) s += A[(long long)r * K + k] * W[k * F + c];
    Out[(long long)r * F + c] = s;
}

// layer-4 transform (F = 1): per-node dot product, WMMA tile would be 15/16 wasted
__global__ void gemv_kernel(const float* __restrict__ H, const float* __restrict__ W,
                            float* __restrict__ Out, int n, int K) {
    int i = blockIdx.x * blockDim.x + threadIdx.x;
    if (i >= n) return;
    const float* row = H + (long long)i * K;
    float s = 0.0f;
#pragma unroll 8
    for (int k = 0; k < K; ++k) s += row[k] * W[k];
    Out[i] = s;
}

// ---------------- self-loop contribution: Out[i,f] = dinv[i]^2 * H[i,f] ----------------

__global__ void self_loop_kernel(const float* __restrict__ dinv,
                                 const float* __restrict__ H,
                                 float* __restrict__ Out, int n, int F) {
    int t = blockIdx.x * blockDim.x + threadIdx.x;
    if (t >= n * F) return;
    int i = t / F;
    float di = dinv[i];
    Out[t] = di * di * H[t];
}

// ---------------- edge scatter: Out[dst,f] += norm[e] * H[src,f] ----------------
// Thread = (edge, feature): at F=32 one wave covers one edge -> coalesced 128B gather
// of H[src][*] and contiguous float atomics on Out[dst][*] resolved in L2.

__global__ void edge_scatter_kernel(const int* __restrict__ src,
                                    const int* __restrict__ dst,
                                    const float* __restrict__ norm,
                                    const float* __restrict__ H,
                                    float* __restrict__ Out,
                                    int E, int F) {
    long long t = (long long)blockIdx.x * blockDim.x + threadIdx.x;
    long long total = (long long)E * F;
    if (t >= total) return;
    int e = (int)(t / F);
    int f = (int)(t - (long long)e * F);
    int s = src[e];
    int d = dst[e];
    atomicAdd(&Out[(long long)d * F + f], H[(long long)s * F + f] * norm[e]);
}

// ---------------- epilogue: bias + activation (0 = relu, 1 = identity, 2 = sigmoid) ----

__global__ void bias_act_kernel(const float* __restrict__ agg,
                                const float* __restrict__ bias,
                                float* __restrict__ Out,
                                int n, int F, int mode) {
    int t = blockIdx.x * blockDim.x + threadIdx.x;
    if (t >= n * F) return;
    int f = t % F;
    float v = agg[t] + bias[f];
    if (mode == 0) {
        v = fmaxf(v, 0.0f);
    } else if (mode == 2) {
        v = 1.0f / (1.0f + __expf(-v));
    }
    Out[t] = v;
}

// ---------------- host-side orchestration ----------------

static inline int cdiv_i(long long a, long long b) { return (int)((a + b - 1) / b); }

extern "C" void kernel_launch(void* const* d_in, const int* in_sizes, int n_in,
                              void* d_out, int out_size, void* d_ws, size_t ws_size,
                              hipStream_t stream) {
    const float* x  = (const float*)d_in[0];
    const int*   ei = (const int*)d_in[1];
    const float* W1 = (const float*)d_in[2];
    const float* b1 = (const float*)d_in[3];
    const float* W2 = (const float*)d_in[4];
    const float* b2 = (const float*)d_in[5];
    const float* W3 = (const float*)d_in[6];
    const float* b3 = (const float*)d_in[7];
    const float* W4 = (const float*)d_in[8];
    const float* b4 = (const float*)d_in[9];

    const int D0 = 54, D1 = 16, D2 = 32, D3 = 32, D4 = 1;
    const int N = in_sizes[0] / D0;   // 100000
    const int E = in_sizes[1] / 2;    // 3200000
    const int* src = ei;
    const int* dst = ei + E;

    float* ws   = (float*)d_ws;
    float* dinv = ws;                                   // N floats
    float* enrm = ws + N;                               // E floats
    float* buf0 = enrm + E;                             // N*32 floats
    float* buf1 = buf0 + (size_t)N * 32;                // N*32 floats
    (void)ws_size; (void)n_in; (void)out_size;

    const int rowTiles = N / 16;      // 6250 full tiles at N=100000
    const int tailRows = N - rowTiles * 16;

    // ---- GCN normalization (with self-loops), edge norms precomputed once ----
    deg_init_kernel<<<cdiv_i(N, TPB), TPB, 0, stream>>>(dinv, N);
    deg_count_kernel<<<cdiv_i(E, TPB), TPB, 0, stream>>>(dst, dinv, E);
    deg_rsqrt_kernel<<<cdiv_i(N, TPB), TPB, 0, stream>>>(dinv, N);
    edge_norm_kernel<<<cdiv_i(E, TPB), TPB, 0, stream>>>(src, dst, dinv, enrm, E);

    auto aggregate = [&](const float* H, float* Out, int F) {
        self_loop_kernel<<<cdiv_i((long long)N * F, TPB), TPB, 0, stream>>>(dinv, H, Out, N, F);
        edge_scatter_kernel<<<cdiv_i((long long)E * F, TPB), TPB, 0, stream>>>(src, dst, enrm, H, Out, E, F);
    };
    auto epilogue = [&](const float* agg, const float* b, float* Out, int F, int mode) {
        bias_act_kernel<<<cdiv_i((long long)N * F, TPB), TPB, 0, stream>>>(agg, b, Out, N, F, mode);
    };
    auto tail = [&](const float* A, const float* W, float* Out, int K, int F) {
        if (tailRows > 0)
            gemm_tail_kernel<<<cdiv_i((long long)tailRows * F, TPB), TPB, 0, stream>>>(
                A, W, Out, rowTiles * 16, N, K, F);
    };

    // ---- Layer 1: 54 -> 16, relu ----
    {
        long long waves = (long long)rowTiles * (D1 / 16);
        gemm_wmma_f32_kernel<D0, D1><<<cdiv_i(waves * 32, TPB), TPB, 0, stream>>>(x, W1, buf0, rowTiles);
        tail(x, W1, buf0, D0, D1);
    }
    aggregate(buf0, buf1, D1);
    epilogue(buf1, b1, buf0, D1, /*relu*/0);

    // ---- Layer 2: 16 -> 32, relu ----
    {
        long long waves = (long long)rowTiles * (D2 / 16);
        gemm_wmma_f32_kernel<D1, D2><<<cdiv_i(waves * 32, TPB), TPB, 0, stream>>>(buf0, W2, buf1, rowTiles);
        tail(buf0, W2, buf1, D1, D2);
    }
    aggregate(buf1, buf0, D2);
    epilogue(buf0, b2, buf1, D2, /*relu*/0);

    // ---- Layer 3: 32 -> 32, linear ----
    {
        long long waves = (long long)rowTiles * (D3 / 16);
        gemm_wmma_f32_kernel<D2, D3><<<cdiv_i(waves * 32, TPB), TPB, 0, stream>>>(buf1, W3, buf0, rowTiles);
        tail(buf1, W3, buf0, D2, D3);
    }
    aggregate(buf0, buf1, D3);
    epilogue(buf1, b3, buf0, D3, /*identity*/1);

    // ---- Layer 4: 32 -> 1, sigmoid ----
    gemv_kernel<<<cdiv_i(N, TPB), TPB, 0, stream>>>(buf0, W4, buf1, N, D3);
    aggregate(buf1, buf0, D4);
    epilogue(buf0, b4, (float*)d_out, D4, /*sigmoid*/2);
}